// FCOS_17832704213392
// MI455X (gfx1250) — compile-verified
//
#include <hip/hip_runtime.h>
#include <hip/hip_bf16.h>
#include <stdint.h>

#define NMS_N   4096
#define IOU_THR 0.5f

// ---------------------------------------------------------------------------
// Kernel 1: box decode + centerness (elementwise over L locations).
// out layout: d_out[0:4096) = keep mask (floats), d_out[4096 + i*5 + k] = pred.
// ---------------------------------------------------------------------------
__global__ void fcos_pred_kernel(const float2* __restrict__ loc,
                                 const float4* __restrict__ deltas,
                                 const int* __restrict__ stride_p,
                                 float* __restrict__ pred, int L) {
  int i = blockIdx.x * blockDim.x + threadIdx.x;
  if (i >= L) return;
  float s = (float)(*stride_p);
  float4 d = deltas[i];
  float2 p = loc[i];
  float c0 = fmaxf(d.x, 0.f), c1 = fmaxf(d.y, 0.f);
  float c2 = fmaxf(d.z, 0.f), c3 = fmaxf(d.w, 0.f);
  float lrmin = fminf(d.x, d.z), tbmin = fminf(d.y, d.w);
  float lrmax = fmaxf(d.x, d.z), tbmax = fmaxf(d.y, d.w);
  float cent = sqrtf((lrmin * tbmin) / (lrmax * tbmax));
  bool allneg = (d.x == -1.f) && (d.y == -1.f) && (d.z == -1.f) && (d.w == -1.f);
  float* o = pred + (size_t)i * 5;
  o[0] = p.x - c0 * s;
  o[1] = p.y - c1 * s;
  o[2] = p.x + c2 * s;
  o[3] = p.y + c3 * s;
  o[4] = allneg ? -1.0f : cent;
}

// ---------------------------------------------------------------------------
// Kernel 2: max over all box coordinates (for the per-class offset).
// ---------------------------------------------------------------------------
__global__ void fcos_maxcoord_kernel(const float* __restrict__ boxes, int n,
                                     float* __restrict__ maxc) {
  __shared__ float red[256];
  float m = -3.0e38f;
  for (int i = threadIdx.x; i < n; i += blockDim.x) m = fmaxf(m, boxes[i]);
  red[threadIdx.x] = m;
  __syncthreads();
  for (int s = 128; s > 0; s >>= 1) {
    if ((int)threadIdx.x < s)
      red[threadIdx.x] = fmaxf(red[threadIdx.x], red[threadIdx.x + s]);
    __syncthreads();
  }
  if (threadIdx.x == 0) *maxc = red[0];
}

// ---------------------------------------------------------------------------
// Kernel 3: single-block bitonic sort of (score, idx) descending (idx asc on
// ties == argmax-first-index order), then gather class-offset boxes in sorted
// order into the workspace.
// ---------------------------------------------------------------------------
__global__ void fcos_sort_kernel(const float* __restrict__ scores,
                                 const float* __restrict__ boxes,
                                 const int* __restrict__ cls,
                                 const float* __restrict__ maxc,
                                 int* __restrict__ s_idx,
                                 float* __restrict__ s_score,
                                 float4* __restrict__ s_boxes, int N) {
  __shared__ float key[NMS_N];
  __shared__ int   idx[NMS_N];
  const int tid = threadIdx.x, NT = blockDim.x;
  for (int i = tid; i < N; i += NT) { key[i] = scores[i]; idx[i] = i; }
  __syncthreads();
  for (int k = 2; k <= N; k <<= 1) {
    for (int j = k >> 1; j > 0; j >>= 1) {
      for (int t = tid; t < N; t += NT) {
        int ixj = t ^ j;
        if (ixj > t) {
          float ka = key[t], kb = key[ixj];
          int   ia = idx[t], ib = idx[ixj];
          bool a_first = (ka > kb) || ((ka == kb) && (ia < ib));
          bool desc = ((t & k) == 0);
          if (desc ? !a_first : a_first) {
            key[t] = kb; key[ixj] = ka; idx[t] = ib; idx[ixj] = ia;
          }
        }
      }
      __syncthreads();
    }
  }
  float off_scale = *maxc + 1.0f;
  for (int i = tid; i < N; i += NT) {
    int j = idx[i];
    s_idx[i] = j;
    s_score[i] = key[i];
    float o = (float)cls[j] * off_scale;
    float4 b;
    b.x = boxes[4 * j + 0] + o;
    b.y = boxes[4 * j + 1] + o;
    b.z = boxes[4 * j + 2] + o;
    b.w = boxes[4 * j + 3] + o;
    s_boxes[i] = b;
  }
}

// ---------------------------------------------------------------------------
// Kernel 4: suppression bit matrix in sorted order. One wave (32 lanes) per
// 32x32 tile. Column boxes are staged into LDS with the CDNA5 async path:
// global_load_async_to_lds_b128 (ASYNCcnt) + s_wait_asynccnt.
// ---------------------------------------------------------------------------
__global__ void fcos_bits_kernel(const float4* __restrict__ sboxes,
                                 unsigned* __restrict__ bits, int N) {
  __shared__ float4 colbox[32];
  const int lane = threadIdx.x;            // 32 threads = one wave32
  const int cblk = blockIdx.x, rblk = blockIdx.y;

  // LDS byte address: generic address of __shared__ truncated to 32 bits is
  // the LDS offset (ISA 10.2 aperture mapping).
  unsigned lds_addr =
      (unsigned)(unsigned long long)(void*)(&colbox[0]) + (unsigned)lane * 16u;
  unsigned long long gaddr =
      (unsigned long long)(const void*)(sboxes + (cblk * 32 + lane));
  // Async DMA one box (16B) per lane into LDS; tracked by ASYNCcnt.
  asm volatile("global_load_async_to_lds_b128 %0, %1, off"
               :: "v"(lds_addr), "v"(gaddr) : "memory");

  // Row box via normal vector load (overlaps with the async copy).
  int row = rblk * 32 + lane;
  float4 rb = sboxes[row];
  float ra = (rb.z - rb.x) * (rb.w - rb.y);

  asm volatile("s_wait_asynccnt 0x0" ::: "memory");
  __syncthreads();

  unsigned word = 0u;
#pragma unroll 4
  for (int jj = 0; jj < 32; ++jj) {
    float4 cb = colbox[jj];
    float ca = (cb.z - cb.x) * (cb.w - cb.y);
    float xi = fminf(rb.z, cb.z) - fmaxf(rb.x, cb.x);
    float yi = fminf(rb.w, cb.w) - fmaxf(rb.y, cb.y);
    float inter = fmaxf(xi, 0.f) * fmaxf(yi, 0.f);
    float iou = inter / (ra + ca - inter);
    if (iou > IOU_THR) word |= (1u << jj);
  }
  bits[(size_t)row * (N / 32) + cblk] = word;
}

// ---------------------------------------------------------------------------
// Kernel 5: serial greedy scan, SINGLE wave32, zero barriers in the loop.
// The 128-word "removed" bitset lives in VGPRs: lane L owns words 4L..4L+3,
// so a kept row is absorbed with one aligned global_load_b128 per lane.
// The alive test is one cross-lane __shfl broadcast (in-order intra-wave).
// Next row is prefetched with global_prefetch while the current one resolves.
// ---------------------------------------------------------------------------
__global__ void fcos_greedy_kernel(const unsigned* __restrict__ bits,
                                   const int* __restrict__ s_idx,
                                   const float* __restrict__ s_score,
                                   float* __restrict__ keep_out, int N) {
  __shared__ int      sidx[NMS_N];
  __shared__ unsigned keep_orig[NMS_N / 32];
  const int lane = threadIdx.x;            // 32 threads = one wave32
  const int W = N / 32;                    // 128 words per row

  for (int p = lane; p < N; p += 32) sidx[p] = s_idx[p];
  for (int w = lane; w < W; w += 32) keep_orig[w] = 0u;

  // Removed bitset in registers; replicate the reference's "scores <
  // first_max" init quirk: positions tied with the global max (other than
  // sorted position 0) are never processed nor kept.
  float maxsc = s_score[0];
  unsigned r[4] = {0u, 0u, 0u, 0u};
  for (int s = 0; s < 4; ++s) {
    int wbase = (lane * 4 + s) * 32;
    unsigned word = 0u;
    for (int b = 0; b < 32; ++b) {
      int p = wbase + b;
      if (p > 0 && s_score[p] == maxsc) word |= (1u << b);
    }
    r[s] = word;
  }
  __syncthreads();   // S_NOP for 1-wave WG; orders LDS init vs. the loop

  for (int i = 0; i < N; ++i) {
    const int w     = i >> 5;        // removed-word index for position i
    const int owner = w >> 2;        // lane holding it
    const int slot  = w & 3;         // uniform across lanes for a given i
    unsigned rw = (slot == 0) ? r[0] : (slot == 1) ? r[1]
                : (slot == 2) ? r[2] : r[3];
    unsigned val = __shfl(rw, owner, 32);        // broadcast owner's word
    bool alive = ((val >> (i & 31)) & 1u) == 0u; // uniform
    if (i + 1 < N)
      __builtin_prefetch(&bits[(size_t)(i + 1) * W + lane * 4], 0, 1);
    if (alive) {
      const uint4 rb = *(const uint4*)&bits[(size_t)i * W + lane * 4];
      r[0] |= rb.x; r[1] |= rb.y; r[2] |= rb.z; r[3] |= rb.w;
      if (lane == 0) {
        int o = sidx[i];
        atomicOr(&keep_orig[o >> 5], 1u << (o & 31));
      }
    }
  }
  __syncthreads();   // compiler-orders lane0's LDS atomics vs. the reads below

  // Emit the full keep mask (also overwrites the 0xAA poison).
  for (int p = lane; p < N; p += 32)
    keep_out[p] = ((keep_orig[p >> 5] >> (p & 31)) & 1u) ? 1.0f : 0.0f;
}

// ---------------------------------------------------------------------------
// Host-side launcher. d_in order: boxes, scores, class_ids, locations,
// deltas, stride. d_out: [keep(4096) | pred(16384x5)] floats.
// ---------------------------------------------------------------------------
extern "C" void kernel_launch(void* const* d_in, const int* in_sizes, int n_in,
                              void* d_out, int out_size, void* d_ws, size_t ws_size,
                              hipStream_t stream) {
  const float*  boxes   = (const float*)d_in[0];
  const float*  scores  = (const float*)d_in[1];
  const int*    cls     = (const int*)d_in[2];
  const float2* loc     = (const float2*)d_in[3];
  const float4* deltas  = (const float4*)d_in[4];
  const int*    strideP = (const int*)d_in[5];

  const int N = in_sizes[1];        // 4096 boxes
  const int L = in_sizes[4] / 4;    // 16384 locations

  float* out  = (float*)d_out;
  float* pred = out + N;

  // Workspace layout (16B-aligned blocks): maxc | s_idx | s_score | s_boxes | bits
  char* w = (char*)d_ws;
  float*    maxc    = (float*)(w);                    // 4 B   (reserve 256)
  int*      s_idx   = (int*)(w + 256);                // 16 KB
  float*    s_score = (float*)(w + 256 + 16384);      // 16 KB
  float4*   s_boxes = (float4*)(w + 33280);           // 64 KB (16B aligned)
  unsigned* bits    = (unsigned*)(w + 98816);         // N*N/8 = 2 MB (16B aligned)
  (void)ws_size; (void)n_in; (void)out_size;

  // Independent elementwise decode.
  fcos_pred_kernel<<<(L + 255) / 256, 256, 0, stream>>>(loc, deltas, strideP,
                                                        pred, L);
  // NMS pipeline (serialized on stream).
  fcos_maxcoord_kernel<<<1, 256, 0, stream>>>(boxes, 4 * N, maxc);
  fcos_sort_kernel<<<1, 1024, 0, stream>>>(scores, boxes, cls, maxc,
                                           s_idx, s_score, s_boxes, N);
  fcos_bits_kernel<<<dim3(N / 32, N / 32), 32, 0, stream>>>(s_boxes, bits, N);
  fcos_greedy_kernel<<<1, 32, 0, stream>>>(bits, s_idx, s_score, out, N);
}